// Prototype2ContrastLoss_54417235640827
// MI455X (gfx1250) — compile-verified
//
#include <hip/hip_runtime.h>
#include <hip/hip_bf16.h>

// Problem constants (from reference setup_inputs)
#define BS 8
#define CCH 256
#define HH 128
#define WW 128
#define NCLS 20
#define HWS (HH * WW)          // 16384
#define NPIX (BS * HWS)        // 131072
#define NCHUNK 64              // pixel chunks for the GEMM
#define PIX_PER_CHUNK (NPIX / NCHUNK)  // 2048 (divides 16384 -> chunk stays in one image)
#define TAU 0.07f
#define EPSN 1e-12f

typedef __attribute__((ext_vector_type(2))) float v2f;
typedef __attribute__((ext_vector_type(4))) float v4f;
typedef __attribute__((ext_vector_type(8))) float v8f;
typedef __attribute__((ext_vector_type(4))) int   v4i;

// ---------------------------------------------------------------------------
// Kernel 0: zero the 32x256 proto accumulator + fg-class bitmask each launch.
// ---------------------------------------------------------------------------
__global__ void proto_zero_kernel(float* __restrict__ protos,
                                  unsigned* __restrict__ fgmask) {
    int i = blockIdx.x * blockDim.x + threadIdx.x;
    if (i < 32 * CCH) protos[i] = 0.0f;
    if (i == 0) *fgmask = 0u;
}

// ---------------------------------------------------------------------------
// Kernel 1: per-pixel inverse L2 norm over channels, background flag,
//           and per-class "exists" bitmask. One thread per pixel; lanes in a
//           wave read consecutive pixels -> fully coalesced 128B streams.
// ---------------------------------------------------------------------------
__global__ void norm_kernel(const float* __restrict__ feat,
                            const int*  __restrict__ gt,
                            float* __restrict__ invn,
                            float* __restrict__ negf,
                            unsigned* __restrict__ fgmask) {
    int P = blockIdx.x * blockDim.x + threadIdx.x;   // 0 .. NPIX-1
    int b = P >> 14;            // / HWS
    int p = P & (HWS - 1);

    const float* fb = feat + (size_t)b * CCH * HWS + p;
    float ss = 0.0f;
#pragma unroll 8
    for (int c = 0; c < CCH; ++c) {
        float v = fb[(size_t)c * HWS];
        ss += v * v;
    }
    invn[P] = 1.0f / fmaxf(sqrtf(ss), EPSN);

    const int* gb = gt + (size_t)b * NCLS * HWS + p;
    int gsum = 0;
    unsigned m = 0u;
#pragma unroll
    for (int k = 0; k < NCLS; ++k) {
        int g = gb[(size_t)k * HWS];
        gsum += g;
        m |= (g == 1) ? (1u << k) : 0u;
    }
    negf[P] = (gsum == 0) ? 1.0f : 0.0f;

    // wave-level OR reduce, one atomic per wave
    for (int off = 16; off > 0; off >>= 1)
        m |= __shfl_xor(m, off, 32);
    if ((threadIdx.x & 31) == 0) atomicOr(fgmask, m);
}

// ---------------------------------------------------------------------------
// Kernel 2: prototypes GEMM via V_WMMA_F32_16X16X4_F32.
//   protos[row, col] = sum_pixels A[row, pix] * featn[pix, col]
//   A rows 0..19 = pred*(gt==1); row 20 = background mask; rows 21..31 = 0.
//
// Macro-step of 8 pixels: the pixel->K-slot assignment is arbitrary (pure
// reduction), so half-wave 0 (K slots 0,1) streams pixels p..p+3 and
// half-wave 1 (K slots 2,3) streams pixels p+4..p+7, each as ONE b128 load
// per stream. WMMA #1 consumes .xy, WMMA #2 consumes .zw.
// All loads are unconditional (valid pointers for every lane); the gt==1 and
// fg/bg selections are value-level cndmasks, so EXEC stays all-ones for the
// convergent WMMA ops and no exec-masked load branches are generated.
// ---------------------------------------------------------------------------
__global__ void proto_gemm_kernel(const float* __restrict__ feat,
                                  const float* __restrict__ pred,
                                  const int*  __restrict__ gt,
                                  const float* __restrict__ invn,
                                  const float* __restrict__ negf,
                                  float* __restrict__ protos) {
    const int lane  = threadIdx.x & 31;
    const int wave  = threadIdx.x >> 5;
    const int gwid  = blockIdx.x * (blockDim.x >> 5) + wave;
    const int ctile = gwid & 15;     // 16 channel tiles of 16
    const int chunk = gwid >> 4;     // 64 pixel chunks
    const int half  = lane >> 4;     // K-slot half: 0 -> K{0,1}, 1 -> K{2,3}
    const int lq    = lane & 15;
    const int c0    = ctile * 16;
    const int hoff  = half * 4;      // pixel offset of this half within macro-step

    const int Pbeg = chunk * PIX_PER_CHUNK;
    const int b    = Pbeg >> 14;          // constant over the whole chunk
    const int p0   = Pbeg & (HWS - 1);

    // per-lane streaming pointers; advance 8 floats (one macro-step) per iter.
    // All 16-byte aligned: p0 multiple of 2048, hoff in {0,4}.
    const float* bfeat = feat + ((size_t)b * CCH + c0 + lq) * HWS + p0 + hoff;
    const float* ppr0  = pred + ((size_t)b * NCLS + lq) * HWS + p0 + hoff;
    const int*   pgt0  = gt   + ((size_t)b * NCLS + lq) * HWS + p0 + hoff;

    const int  mm    = lq + 16;
    const bool isFg1 = (mm < NCLS);
    const bool isBg  = (mm == NCLS);
    const int  m1    = isFg1 ? mm : 0;    // valid pointer even when unused
    const float* ppr1 = pred + ((size_t)b * NCLS + m1) * HWS + p0 + hoff;
    const int*   pgt1 = gt   + ((size_t)b * NCLS + m1) * HWS + p0 + hoff;
    const float* pinv = invn + Pbeg + hoff;
    const float* pneg = negf + Pbeg + hoff;

    v8f acc0 = {};   // classes 0..15
    v8f acc1 = {};   // classes 16..31 (20 = background)

    for (int it = 0; it < PIX_PER_CHUNK / 8; ++it) {
        // one b128 load per stream per macro-step
        v4f f   = *(const v4f*)bfeat;
        v4f iv  = *(const v4f*)pinv;
        v4f pr0 = *(const v4f*)ppr0;
        v4i g0  = *(const v4i*)pgt0;
        v4f pr1 = *(const v4f*)ppr1;
        v4i g1  = *(const v4i*)pgt1;
        v4f ngv = *(const v4f*)pneg;

        // B = featn (feat * per-pixel inverse norm)
        v2f bva; bva.x = f.x * iv.x; bva.y = f.y * iv.y;
        v2f bvb; bvb.x = f.z * iv.z; bvb.y = f.w * iv.w;

        // A tile 0: classes 0..15  (value-level selects -> v_cndmask)
        v2f a0a; a0a.x = (g0.x == 1) ? pr0.x : 0.0f;
                 a0a.y = (g0.y == 1) ? pr0.y : 0.0f;
        v2f a0b; a0b.x = (g0.z == 1) ? pr0.z : 0.0f;
                 a0b.y = (g0.w == 1) ? pr0.w : 0.0f;

        // A tile 1: classes 16..19 fg, 20 bg, 21..31 zero (branch-free)
        float fx = (g1.x == 1) ? pr1.x : 0.0f;
        float fy = (g1.y == 1) ? pr1.y : 0.0f;
        float fz = (g1.z == 1) ? pr1.z : 0.0f;
        float fw = (g1.w == 1) ? pr1.w : 0.0f;
        v2f a1a; a1a.x = isFg1 ? fx : (isBg ? ngv.x : 0.0f);
                 a1a.y = isFg1 ? fy : (isBg ? ngv.y : 0.0f);
        v2f a1b; a1b.x = isFg1 ? fz : (isBg ? ngv.z : 0.0f);
                 a1b.y = isFg1 ? fw : (isBg ? ngv.w : 0.0f);

        acc0 = __builtin_amdgcn_wmma_f32_16x16x4_f32(false, a0a, false, bva,
                                                     (short)0, acc0, false, false);
        acc1 = __builtin_amdgcn_wmma_f32_16x16x4_f32(false, a1a, false, bva,
                                                     (short)0, acc1, false, false);
        acc0 = __builtin_amdgcn_wmma_f32_16x16x4_f32(false, a0b, false, bvb,
                                                     (short)0, acc0, false, false);
        acc1 = __builtin_amdgcn_wmma_f32_16x16x4_f32(false, a1b, false, bvb,
                                                     (short)0, acc1, false, false);

        bfeat += 8; ppr0 += 8; pgt0 += 8; ppr1 += 8; pgt1 += 8;
        pinv  += 8; pneg += 8;
    }

    // D layout: lanes 0-15 -> N=lane, M=i ; lanes 16-31 -> N=lane-16, M=8+i
    const int col = c0 + lq;
#pragma unroll
    for (int i = 0; i < 8; ++i) {
        int row = i + half * 8;
        atomicAdd(&protos[row * CCH + col], acc0[i]);
        atomicAdd(&protos[(row + 16) * CCH + col], acc1[i]);
    }
}

// ---------------------------------------------------------------------------
// Kernel 3: tiny finalize. Normalize the 21 prototypes, build the 21x21
// exp-similarity matrix, column sums, and the scalar loss (replicating the
// reference's stale-index background term E[19][20]).
// ---------------------------------------------------------------------------
__global__ void finalize_kernel(const float* __restrict__ protos,
                                const unsigned* __restrict__ fgmask_p,
                                float* __restrict__ out) {
    __shared__ float pn[21 * CCH];
    __shared__ float rinv[21];
    __shared__ float Els[21 * 21];
    __shared__ float colsum[21];

    const int tid  = threadIdx.x;          // 256 threads
    const int wave = tid >> 5;
    const int lane = tid & 31;

    for (int i = tid; i < 21 * CCH; i += 256) pn[i] = protos[i];
    __syncthreads();

    for (int r = wave; r < 21; r += 8) {
        float s = 0.0f;
        for (int c = lane; c < CCH; c += 32) {
            float v = pn[r * CCH + c];
            s += v * v;
        }
        for (int off = 16; off > 0; off >>= 1)
            s += __shfl_xor(s, off, 32);
        if (lane == 0) rinv[r] = 1.0f / fmaxf(sqrtf(s), EPSN);
    }
    __syncthreads();

    for (int idx = tid; idx < 21 * 21; idx += 256) {
        int i = idx / 21, j = idx % 21;
        float d = 0.0f;
        for (int c = 0; c < CCH; ++c) d += pn[i * CCH + c] * pn[j * CCH + c];
        d *= rinv[i] * rinv[j];
        Els[idx] = expf(d / TAU);
    }
    __syncthreads();

    if (tid < 21) {
        float s = 0.0f;
        for (int i = 0; i < 21; ++i) s += Els[i * 21 + tid];
        colsum[tid] = s;
    }
    __syncthreads();

    if (tid == 0) {
        unsigned m = *fgmask_p;
        float fg_loss = 0.0f, cnt = 0.0f;
        for (int k = 0; k < NCLS; ++k) {
            if (m & (1u << k)) {
                fg_loss += -logf(Els[k * 21 + k] / colsum[k]);
                cnt += 1.0f;
            }
        }
        // reference bug replication: uses E[K-1][K] with K=20 -> E[19][20]
        float bg_loss = -logf(Els[19 * 21 + 20] / colsum[20]);
        out[0] = (fg_loss + bg_loss) / (cnt + 1.0f);
    }
}

// ---------------------------------------------------------------------------
extern "C" void kernel_launch(void* const* d_in, const int* in_sizes, int n_in,
                              void* d_out, int out_size, void* d_ws, size_t ws_size,
                              hipStream_t stream) {
    const float* feat = (const float*)d_in[0];
    const float* pred = (const float*)d_in[1];
    const int*   gt   = (const int*)d_in[2];
    float* out = (float*)d_out;

    // workspace layout (floats): protos[32*256] | fgmask | pad | invn[NPIX] | negf[NPIX]
    float* ws      = (float*)d_ws;
    float* protos  = ws;                       // 8192 floats
    unsigned* fgm  = (unsigned*)(ws + 8192);   // 1 dword
    float* invn    = ws + 8200;                // NPIX floats
    float* negf    = invn + NPIX;              // NPIX floats

    proto_zero_kernel<<<(32 * CCH + 255) / 256, 256, 0, stream>>>(protos, fgm);
    norm_kernel<<<NPIX / 256, 256, 0, stream>>>(feat, gt, invn, negf, fgm);
    // 1024 waves = 16 channel tiles x 64 chunks; 8 waves per 256-thread block.
    // Consecutive wave ids cover all 16 ctiles of one chunk -> pred/gt slices
    // stay L2-resident across the redundant reads.
    proto_gemm_kernel<<<(16 * NCHUNK) / 8, 256, 0, stream>>>(feat, pred, gt,
                                                             invn, negf, protos);
    finalize_kernel<<<1, 256, 0, stream>>>(protos, fgm, out);
}